// BiLinearAttention_71339406786718
// MI455X (gfx1250) — compile-verified
//
#include <hip/hip_runtime.h>

typedef float v2f __attribute__((ext_vector_type(2)));
typedef float v8f __attribute__((ext_vector_type(8)));
typedef unsigned int u32;
typedef u32 u32x4 __attribute__((ext_vector_type(4)));
typedef int  i32x4 __attribute__((ext_vector_type(4)));
typedef int  i32x8 __attribute__((ext_vector_type(8)));

#define B_ 8
#define N_ 128
#define K_ 64
#define D_ 256
#define H_ 8
#define NPAIR (B_ * N_)   // 1024
#define HD    (H_ * D_)   // 2048
#define LSTR  260         // LDS row stride (TDM pad: 256 + 4 dwords) -> no bank conflicts

// ---------------------------------------------------------------------------
// TDM: DMA a 2D f32 tile (rows x 256, row-major, stride 256) from global
// memory into LDS, padding each 256-dword row with 4 dwords (LDS stride 260).
// D# layout per CDNA5 ISA.
// ---------------------------------------------------------------------------
__device__ __forceinline__ void tdm_load_rows256(const void* gptr, u32 lds_off,
                                                 u32 rows) {
  const unsigned long long ga = (unsigned long long)(size_t)gptr;
  u32x4 g0;
  g0.x = 1u;                                    // count=1 (valid), user mode
  g0.y = lds_off;                               // lds_addr (bytes)
  g0.z = (u32)(ga & 0xFFFFFFFFu);               // global_addr[31:0]
  g0.w = (u32)((ga >> 32) & 0x01FFFFFFu)        // global_addr[56:32]
       | (2u << 30);                            // type = 2 ("image")
  i32x8 g1;
  g1[0] = (int)((2u << 16)                      // data_size = 4 bytes
              | (1u << 20)                      // pad_enable
              | (7u << 22)                      // pad_interval: 256 dwords
              | (3u << 25));                    // pad_amount: 4 dwords
  g1[1] = (int)(256u << 16);                    // tensor_dim0 = 256
  g1[2] = (int)(rows << 16);                    // tensor_dim1 = rows
  g1[3] = (int)(256u << 16);                    // tile_dim0 = 256
  g1[4] = (int)rows;                            // tile_dim1 = rows, tile_dim2 = 0
  g1[5] = 256;                                  // tensor_dim0_stride = 256
  g1[6] = 0;                                    // tensor_dim1_stride = 0 (2D)
  g1[7] = 0;
  i32x4 gz = {0, 0, 0, 0};                      // groups 2/3 unused (2D tile)
#if defined(__clang_major__) && __clang_major__ >= 23
  i32x8 gz8 = {0, 0, 0, 0, 0, 0, 0, 0};
  __builtin_amdgcn_tensor_load_to_lds(g0, g1, gz, gz, gz8, 0);
#else
  __builtin_amdgcn_tensor_load_to_lds(g0, g1, gz, gz, 0);
#endif
}

__device__ __forceinline__ u32 lds_offset(const void* p) {
  return (u32)(size_t)p;   // generic LDS pointer: low 32 bits = wave-relative offset
}

// ---------------------------------------------------------------------------
// Kernel 1: T[p][h*D+e] = sum_d q[p][d] * W[h][d][e]
// Grid (32, 16), block 128 (4 waves). Block carries 64 pairs (4 pair-groups):
// one W fetch feeds 4 wmma (4 accumulators), cutting W L2 traffic 4x.
// q tile (64x256) staged via TDM with padded stride 260 (bank-conflict-free).
// ---------------------------------------------------------------------------
__global__ __launch_bounds__(128)
void bla_qw_gemm(const float* __restrict__ q, const float* __restrict__ W,
                 float* __restrict__ T) {
  __shared__ float qs[64 * LSTR];
  const int tid = threadIdx.x;
  const int pairbase = blockIdx.y * 64;

  if (tid < 32) {                                // wave 0 issues the DMA
    tdm_load_rows256(q + (size_t)pairbase * D_, lds_offset(qs), 64u);
    __builtin_amdgcn_s_wait_tensorcnt(0);
  }
  __syncthreads();

  const int lane  = tid & 31;
  const int col   = lane & 15;
  const int half  = lane >> 4;
  const int ntile = blockIdx.x * 4 + (tid >> 5);   // 0..127
  const int h     = ntile >> 4;
  const int ebase = (ntile & 15) << 4;

  const float* wp = W + (size_t)h * (D_ * D_) + ebase + col;
  const float* a0p = &qs[(0 * 16 + col) * LSTR + 2 * half];
  const float* a1p = &qs[(1 * 16 + col) * LSTR + 2 * half];
  const float* a2p = &qs[(2 * 16 + col) * LSTR + 2 * half];
  const float* a3p = &qs[(3 * 16 + col) * LSTR + 2 * half];

  v8f c0 = {0.f,0.f,0.f,0.f,0.f,0.f,0.f,0.f};
  v8f c1 = c0, c2 = c0, c3 = c0;
  #pragma unroll 4
  for (int s = 0; s < 64; ++s) {
    const int d0 = 4 * s;
    v2f bm;                                      // B (4x16): W rows d0+2h, +1
    bm.x = wp[(size_t)(d0 + 2 * half) * D_];
    bm.y = wp[(size_t)(d0 + 2 * half + 1) * D_];
    v2f a0 = *(const v2f*)(a0p + d0);            // ds_load_b64, conflict-free
    v2f a1 = *(const v2f*)(a1p + d0);
    v2f a2 = *(const v2f*)(a2p + d0);
    v2f a3 = *(const v2f*)(a3p + d0);
    c0 = __builtin_amdgcn_wmma_f32_16x16x4_f32(false, a0, false, bm, (short)0, c0, false, false);
    c1 = __builtin_amdgcn_wmma_f32_16x16x4_f32(false, a1, false, bm, (short)0, c1, false, false);
    c2 = __builtin_amdgcn_wmma_f32_16x16x4_f32(false, a2, false, bm, (short)0, c2, false, false);
    c3 = __builtin_amdgcn_wmma_f32_16x16x4_f32(false, a3, false, bm, (short)0, c3, false, false);
  }

  #pragma unroll
  for (int pg = 0; pg < 4; ++pg) {
    const v8f c = (pg == 0) ? c0 : (pg == 1) ? c1 : (pg == 2) ? c2 : c3;
    float* tp = T + (size_t)(pairbase + pg * 16 + 8 * half) * HD + ntile * 16 + col;
    #pragma unroll
    for (int r = 0; r < 8; ++r) tp[(size_t)r * HD] = c[r];
  }
}

// ---------------------------------------------------------------------------
// Kernel 2: per (b,n) pair: scores = k @ T^T + b, flat-reshaped softmax
// (reference reshape (B,N,K,H)->(B,N,H,K) is a reinterpretation), attn out,
// out = (sum_h' attn) @ v. Grid 1024, block 128 (4 waves).
// k[pair] (64x256) and T row (8x256) staged via TDM, padded stride 260.
// ---------------------------------------------------------------------------
__global__ __launch_bounds__(128)
void bla_attn(const float* __restrict__ k, const float* __restrict__ v,
              const float* __restrict__ bias, const float* __restrict__ T,
              float* __restrict__ out, float* __restrict__ attn) {
  __shared__ float ks[64 * LSTR];
  __shared__ float Ts[8 * LSTR];
  __shared__ float Sf[512];       // scores flat f = kk*8 + h (== (K,H) C-order)
  __shared__ float Af[512];       // attn flat (same order)
  __shared__ float asum[64];
  __shared__ float gm[8], gi[8];
  const int tid  = threadIdx.x;
  const int pair = blockIdx.x;

  if (tid < 32) {                                // wave 0 issues both DMAs
    tdm_load_rows256(k + (size_t)pair * (K_ * D_), lds_offset(ks), 64u);
    tdm_load_rows256(T + (size_t)pair * HD, lds_offset(Ts), 8u);
    __builtin_amdgcn_s_wait_tensorcnt(0);
  }
  __syncthreads();

  const int lane = tid & 31, wave = tid >> 5;
  const int col  = lane & 15, half = lane >> 4;

  // A = k rows (M = kk in [16*wave,16*wave+16)), B = T^T (K=d, N=h; cols 8..15 junk)
  const float* krow = &ks[(wave * 16 + col) * LSTR + 2 * half];
  const float* trow = &Ts[(col & 7) * LSTR + 2 * half];

  v8f c = {0.f,0.f,0.f,0.f,0.f,0.f,0.f,0.f};
  #pragma unroll 4
  for (int s = 0; s < 64; ++s) {
    const int d0 = 4 * s;
    v2f a  = *(const v2f*)(krow + d0);           // ds_load_b64
    v2f bm = *(const v2f*)(trow + d0);           // ds_load_b64
    c = __builtin_amdgcn_wmma_f32_16x16x4_f32(false, a, false, bm, (short)0, c,
                                              false, false);
  }

  if (col < 8) {                                 // store valid h cols + bias
    const float bb = bias[col];
    const int kkbase = wave * 16 + 8 * half;
    #pragma unroll
    for (int r = 0; r < 8; ++r) Sf[(kkbase + r) * 8 + col] = c[r] + bb;
  }
  __syncthreads();

  if (tid < 8) {                                 // softmax stats per flat 64-group
    const float* g = &Sf[tid * 64];
    float m = -3.402823466e+38f;
    for (int i = 0; i < 64; ++i) m = fmaxf(m, g[i]);
    float s = 0.f;
    for (int i = 0; i < 64; ++i) s += __expf(g[i] - m);
    gm[tid] = m;
    gi[tid] = 1.0f / s;
  }
  __syncthreads();

  {
    const int f0 = tid * 4;
    const int g  = f0 >> 6;
    const float m = gm[g], inv = gi[g];
    float4 av;
    av.x = __expf(Sf[f0 + 0] - m) * inv;
    av.y = __expf(Sf[f0 + 1] - m) * inv;
    av.z = __expf(Sf[f0 + 2] - m) * inv;
    av.w = __expf(Sf[f0 + 3] - m) * inv;
    *(float4*)(attn + (size_t)pair * 512 + f0) = av;   // coalesced b128
    Af[f0 + 0] = av.x; Af[f0 + 1] = av.y; Af[f0 + 2] = av.z; Af[f0 + 3] = av.w;
  }
  __syncthreads();

  if (tid < 64) {
    float s = 0.f;
    #pragma unroll
    for (int g = 0; g < 8; ++g) s += Af[g * 64 + tid];
    asum[tid] = s;
  }
  __syncthreads();

  {                                              // out = asum . v, v streamed once
    const int d0 = tid * 2;
    float acc0 = 0.f, acc1 = 0.f;
    const float* vp = v + (size_t)pair * (K_ * D_) + d0;
    #pragma unroll 4
    for (int kk = 0; kk < 64; ++kk) {
      const float a = asum[kk];
      v2f vv = *(const v2f*)(vp + (size_t)kk * D_);      // coalesced b64
      acc0 += a * vv.x;
      acc1 += a * vv.y;
    }
    v2f r; r.x = acc0; r.y = acc1;
    *(v2f*)(out + (size_t)pair * D_ + d0) = r;
  }
}

extern "C" void kernel_launch(void* const* d_in, const int* in_sizes, int n_in,
                              void* d_out, int out_size, void* d_ws, size_t ws_size,
                              hipStream_t stream) {
  (void)in_sizes; (void)n_in; (void)out_size; (void)ws_size;
  const float* q    = (const float*)d_in[0];
  const float* k    = (const float*)d_in[1];
  const float* v    = (const float*)d_in[2];
  const float* W    = (const float*)d_in[3];
  const float* bias = (const float*)d_in[4];
  float* out  = (float*)d_out;
  float* attn = out + (size_t)NPAIR * D_;   // outputs concatenated flat
  float* T    = (float*)d_ws;               // needs NPAIR*HD*4 = 8 MiB scratch

  bla_qw_gemm<<<dim3(HD / 64, NPAIR / 64), dim3(128), 0, stream>>>(q, W, T);
  bla_attn<<<dim3(NPAIR), dim3(128), 0, stream>>>(k, v, bias, T, out, attn);
}